// GAT_20100446945619
// MI455X (gfx1250) — compile-verified
//
#include <hip/hip_runtime.h>

typedef float v2f __attribute__((ext_vector_type(2)));
typedef float v8f __attribute__((ext_vector_type(8)));

#define N_NODES 50000
#define E_EDGES 800000
#define E_TOT   850000     // + self loops
#define G_GRAPHS 512
#define IN_DIM  64
#define HC      128
#define NHEAD   2
#define NEG_SLOPE 0.2f

// ---- monotonic float<->uint ordering key (for atomic segment-max) ----
__device__ __forceinline__ unsigned fkey(float f) {
    unsigned b = __float_as_uint(f);
    return (b & 0x80000000u) ? ~b : (b | 0x80000000u);
}
__device__ __forceinline__ float fdecode(unsigned k) {
    return (k & 0x80000000u) ? __uint_as_float(k ^ 0x80000000u)
                             : __uint_as_float(~k);
}

// ---------------- utility: zero buffers ----------------
__global__ void zero_f32(float* p, int n) {
    int i = blockIdx.x * blockDim.x + threadIdx.x;
    if (i < n) p[i] = 0.0f;
}
__global__ void zero_u32(unsigned* p, int n) {
    int i = blockIdx.x * blockDim.x + threadIdx.x;
    if (i < n) p[i] = 0u;
}

// ---------------- GEMM + bias via V_WMMA_F32_16X16X4_F32 ----------------
// Y[M,NC] = X[M,K] @ W[K,NC] + bias ;  one wave per 16x16 output tile.
// A (16x4 f32): lanes 0-15 -> M rows, v.x/v.y = K {0,1}; lanes 16-31 -> K {2,3}.
// B (4x16 f32): lanes = N cols, mirrored K layout.
// C/D: VGPR j, lanes 0-15: M=j, N=lane; lanes 16-31: M=j+8, N=lane-16.
__global__ void gemm_bias_wmma(const float* __restrict__ X,
                               const float* __restrict__ W,
                               const float* __restrict__ bias,
                               float* __restrict__ Y,
                               int K, int NC) {
    const int lane = threadIdx.x;                 // 0..31 (wave32)
    const int tm   = blockIdx.x;                  // row tile
    const int tn   = blockIdx.y * blockDim.y + threadIdx.y; // col tile
    const int half = lane >> 4;
    const int l16  = lane & 15;

    const int row = tm * 16 + l16;
    const int col = tn * 16 + l16;
    const float* xrow = X + (size_t)row * K;

    v8f acc = {0.f, 0.f, 0.f, 0.f, 0.f, 0.f, 0.f, 0.f};

    #pragma unroll 4
    for (int k = 0; k < K; k += 4) {
        const int ka = k + half * 2;
        v2f A; A.x = xrow[ka];                    A.y = xrow[ka + 1];
        v2f B; B.x = W[(size_t)ka * NC + col];    B.y = W[(size_t)(ka + 1) * NC + col];
        acc = __builtin_amdgcn_wmma_f32_16x16x4_f32(
                  false, A, false, B, (short)0, acc, false, false);
    }

    const float bv = bias[col];
    #pragma unroll
    for (int j = 0; j < 8; ++j) {
        const int orow = tm * 16 + j + half * 8;
        Y[(size_t)orow * NC + col] = acc[j] + bv;
    }
}

// ---------------- per-edge src/dst helper ----------------
__device__ __forceinline__ void edge_sd(const int* __restrict__ edge, int e,
                                        int& s, int& d) {
    if (e < E_EDGES) { s = edge[e]; d = edge[E_EDGES + e]; }
    else             { s = d = e - E_EDGES; }   // self loops appended
}

// ---------------- edge score + atomic segment-max ----------------
// one wave per edge; lane covers 4 channels (lanes 0-15 head0, 16-31 head1)
__global__ void edge_score(const float* __restrict__ xl,
                           const float* __restrict__ xr,
                           const int*   __restrict__ edge,
                           const float* __restrict__ att,   // [H*C] = 128
                           float*       __restrict__ score, // [E_TOT*2]
                           unsigned*    __restrict__ smax)  // [N*2] keys
{
    const int wid  = blockIdx.x * (blockDim.x >> 5) + (threadIdx.x >> 5);
    const int lane = threadIdx.x & 31;
    if (wid >= E_TOT) return;

    int s, d; edge_sd(edge, wid, s, d);
    const int ch = lane * 4;

    const float4 a = *(const float4*)(xl + (size_t)s * HC + ch);
    const float4 b = *(const float4*)(xr + (size_t)d * HC + ch);
    const float4 w = *(const float4*)(att + ch);

    float p = 0.f, v;
    v = a.x + b.x; v = v > 0.f ? v : NEG_SLOPE * v; p += v * w.x;
    v = a.y + b.y; v = v > 0.f ? v : NEG_SLOPE * v; p += v * w.y;
    v = a.z + b.z; v = v > 0.f ? v : NEG_SLOPE * v; p += v * w.z;
    v = a.w + b.w; v = v > 0.f ? v : NEG_SLOPE * v; p += v * w.w;

    // reduce within each 16-lane head group
    p += __shfl_xor(p, 1);
    p += __shfl_xor(p, 2);
    p += __shfl_xor(p, 4);
    p += __shfl_xor(p, 8);

    if ((lane & 15) == 0) {
        const int h = lane >> 4;
        score[(size_t)wid * 2 + h] = p;
        atomicMax(&smax[(size_t)d * 2 + h], fkey(p));
    }
}

// ---------------- exp(score - segmax) + atomic denominator ----------------
__global__ void edge_exp(const int* __restrict__ edge,
                         float* __restrict__ score,        // in: score, out: exp
                         const unsigned* __restrict__ smax,
                         float* __restrict__ denom)        // [N*2]
{
    const int idx = blockIdx.x * blockDim.x + threadIdx.x;
    if (idx >= E_TOT * 2) return;
    const int e = idx >> 1, h = idx & 1;
    int s, d; edge_sd(edge, e, s, d); (void)s;
    const float m  = fdecode(smax[(size_t)d * 2 + h]);
    const float ex = __expf(score[idx] - m);
    score[idx] = ex;
    atomicAdd(&denom[(size_t)d * 2 + h], ex);
}

// ---------------- weighted scatter-add: out[dst] += xl[src] * alpha ----------------
__global__ void edge_agg(const float* __restrict__ xl,
                         const float* __restrict__ ex,     // [E_TOT*2]
                         const float* __restrict__ denom,  // [N*2]
                         const int*   __restrict__ edge,
                         float*       __restrict__ out)    // [N*HC] pre-zeroed
{
    const int wid  = blockIdx.x * (blockDim.x >> 5) + (threadIdx.x >> 5);
    const int lane = threadIdx.x & 31;
    if (wid >= E_TOT) return;

    int s, d; edge_sd(edge, wid, s, d);
    const int h  = lane >> 4;
    const int ch = lane * 4;

    const float alpha = ex[(size_t)wid * 2 + h] /
                        (denom[(size_t)d * 2 + h] + 1e-16f);
    const float4 v = *(const float4*)(xl + (size_t)s * HC + ch);
    float* o = out + (size_t)d * HC + ch;
    atomicAdd(o + 0, v.x * alpha);
    atomicAdd(o + 1, v.y * alpha);
    atomicAdd(o + 2, v.z * alpha);
    atomicAdd(o + 3, v.w * alpha);
}

// ---------------- in-place bias + ReLU ----------------
__global__ void bias_relu(float* __restrict__ y, const float* __restrict__ bias, int n) {
    const int i = blockIdx.x * blockDim.x + threadIdx.x;
    if (i < n) {
        float v = y[i] + bias[i & (HC - 1)];
        y[i] = v > 0.f ? v : 0.f;
    }
}

// ---------------- global add pool ----------------
__global__ void pool_add(const float* __restrict__ h,
                         const int*   __restrict__ batch,
                         float*       __restrict__ pooled) // [G*HC] pre-zeroed
{
    const int i = blockIdx.x * blockDim.x + threadIdx.x;
    if (i >= N_NODES * HC) return;
    const int node = i >> 7, c = i & (HC - 1);
    atomicAdd(&pooled[(size_t)batch[node] * HC + c], h[i]);
}

// ---------------- final linear head: out[g] = pooled[g] . Wout + bout ----------------
__global__ void final_out(const float* __restrict__ pooled,
                          const float* __restrict__ Wout,  // [HC]
                          const float* __restrict__ bout,  // [1]
                          float*       __restrict__ out)   // [G]
{
    const int g    = blockIdx.x * (blockDim.x >> 5) + (threadIdx.x >> 5);
    const int lane = threadIdx.x & 31;
    if (g >= G_GRAPHS) return;
    const int ch = lane * 4;
    const float4 p = *(const float4*)(pooled + (size_t)g * HC + ch);
    const float4 w = *(const float4*)(Wout + ch);
    float t = p.x * w.x + p.y * w.y + p.z * w.z + p.w * w.w;
    t += __shfl_xor(t, 1);
    t += __shfl_xor(t, 2);
    t += __shfl_xor(t, 4);
    t += __shfl_xor(t, 8);
    t += __shfl_xor(t, 16);
    if (lane == 0) out[g] = t + bout[0];
}

// =====================================================================
extern "C" void kernel_launch(void* const* d_in, const int* in_sizes, int n_in,
                              void* d_out, int out_size, void* d_ws, size_t ws_size,
                              hipStream_t stream) {
    const float* x     = (const float*)d_in[0];
    const int*   edge  = (const int*)  d_in[1];
    const int*   batch = (const int*)  d_in[2];
    const float* W1l = (const float*)d_in[3];
    const float* b1l = (const float*)d_in[4];
    const float* W1r = (const float*)d_in[5];
    const float* b1r = (const float*)d_in[6];
    const float* att1  = (const float*)d_in[7];
    const float* bias1 = (const float*)d_in[8];
    const float* W2l = (const float*)d_in[9];
    const float* b2l = (const float*)d_in[10];
    const float* W2r = (const float*)d_in[11];
    const float* b2r = (const float*)d_in[12];
    const float* att2  = (const float*)d_in[13];
    const float* bias2 = (const float*)d_in[14];
    const float* Wout  = (const float*)d_in[15];
    const float* bout  = (const float*)d_in[16];
    float* out = (float*)d_out;

    // workspace layout
    float*    P0   = (float*)d_ws;                         // N*HC (xl)
    float*    P1   = P0 + (size_t)N_NODES * HC;            // N*HC (xr1 / layer1 out)
    float*    P2   = P1 + (size_t)N_NODES * HC;            // N*HC (xr2 / layer2 out)
    float*    EX   = P2 + (size_t)N_NODES * HC;            // E_TOT*2
    unsigned* SMAX = (unsigned*)(EX + (size_t)E_TOT * 2);  // N*2
    float*    DEN  = (float*)(SMAX + (size_t)N_NODES * 2); // N*2
    float*    POOL = DEN + (size_t)N_NODES * 2;            // G*HC

    const dim3 gblk(32, 4);
    const dim3 ggrd(N_NODES / 16, (HC / 16) / 4);          // (3125, 2)
    const int  ewaves = 256 / 32;
    const dim3 egrd((E_TOT + ewaves - 1) / ewaves);
    const int  th = 256;

    // ---------------- layer 1 ----------------
    gemm_bias_wmma<<<ggrd, gblk, 0, stream>>>(x, W1l, b1l, P0, IN_DIM, HC);
    gemm_bias_wmma<<<ggrd, gblk, 0, stream>>>(x, W1r, b1r, P1, IN_DIM, HC);
    zero_u32<<<(N_NODES * 2 + th - 1) / th, th, 0, stream>>>(SMAX, N_NODES * 2);
    zero_f32<<<(N_NODES * 2 + th - 1) / th, th, 0, stream>>>(DEN, N_NODES * 2);
    edge_score<<<egrd, th, 0, stream>>>(P0, P1, edge, att1, EX, SMAX);
    zero_f32<<<((N_NODES * HC) + th - 1) / th, th, 0, stream>>>(P1, N_NODES * HC);
    edge_exp<<<((E_TOT * 2) + th - 1) / th, th, 0, stream>>>(edge, EX, SMAX, DEN);
    edge_agg<<<egrd, th, 0, stream>>>(P0, EX, DEN, edge, P1);
    bias_relu<<<((N_NODES * HC) + th - 1) / th, th, 0, stream>>>(P1, bias1, N_NODES * HC);

    // ---------------- layer 2 (input = P1) ----------------
    gemm_bias_wmma<<<ggrd, gblk, 0, stream>>>(P1, W2l, b2l, P0, HC, HC);
    gemm_bias_wmma<<<ggrd, gblk, 0, stream>>>(P1, W2r, b2r, P2, HC, HC);
    zero_u32<<<(N_NODES * 2 + th - 1) / th, th, 0, stream>>>(SMAX, N_NODES * 2);
    zero_f32<<<(N_NODES * 2 + th - 1) / th, th, 0, stream>>>(DEN, N_NODES * 2);
    edge_score<<<egrd, th, 0, stream>>>(P0, P2, edge, att2, EX, SMAX);
    zero_f32<<<((N_NODES * HC) + th - 1) / th, th, 0, stream>>>(P2, N_NODES * HC);
    edge_exp<<<((E_TOT * 2) + th - 1) / th, th, 0, stream>>>(edge, EX, SMAX, DEN);
    edge_agg<<<egrd, th, 0, stream>>>(P0, EX, DEN, edge, P2);
    bias_relu<<<((N_NODES * HC) + th - 1) / th, th, 0, stream>>>(P2, bias2, N_NODES * HC);

    // ---------------- pooling + head ----------------
    zero_f32<<<(G_GRAPHS * HC + th - 1) / th, th, 0, stream>>>(POOL, G_GRAPHS * HC);
    pool_add<<<((N_NODES * HC) + th - 1) / th, th, 0, stream>>>(P2, batch, POOL);
    final_out<<<(G_GRAPHS + ewaves - 1) / ewaves, th, 0, stream>>>(POOL, Wout, bout, out);
}